// PoseFeatureNet_21749714387316
// MI455X (gfx1250) — compile-verified
//
#include <hip/hip_runtime.h>
#include <hip/hip_bf16.h>
#include <cstdint>

// ---------------------------------------------------------------------------
// CDNA5 (gfx1250) fp32 WMMA helpers: V_WMMA_F32_16X16X4_F32
//   A: 16x4 fp32  -> 2 VGPRs (lane%16 = M row; lanes>=16 hold K+2)
//   B: 4x16 fp32  -> 2 VGPRs (lane%16 = N col; lanes>=16 hold K+2)
//   C/D: 16x16    -> 8 VGPRs (row = q + 8*(lane>=16), col = lane%16)
// ---------------------------------------------------------------------------
typedef __attribute__((ext_vector_type(2))) float v2f;
typedef __attribute__((ext_vector_type(8))) float v8f;

__device__ __forceinline__ v8f wmma_f32_16x16x4(v2f a, v2f b, v8f c) {
  // 8 args: (neg_a, A, neg_b, B, c_mod, C, reuse_a, reuse_b)
  return __builtin_amdgcn_wmma_f32_16x16x4_f32(false, a, false, b, (short)0, c,
                                               false, false);
}

__device__ __forceinline__ v8f zero8() {
  v8f z;
#pragma unroll
  for (int i = 0; i < 8; ++i) z[i] = 0.0f;
  return z;
}

__device__ __forceinline__ float sigmf(float x) {
  return 1.0f / (1.0f + __expf(-x));
}

// ---------------------------------------------------------------------------
// CDNA5 async global->LDS copy (ASYNCcnt-tracked), guarded so the build never
// breaks: falls back to a synchronous float4 copy when the builtin is absent.
// Probe result (round 2 diagnostic): the builtin takes pointers to a 16-byte
// int vector, global source in AS(1), LDS destination in AS(3).
// ---------------------------------------------------------------------------
#if defined(__AMDGCN__) && __has_builtin(__builtin_amdgcn_global_load_async_to_lds_b128)
#define HAVE_ASYNC_LDS 1
#else
#define HAVE_ASYNC_LDS 0
#endif

typedef int v4i_vs __attribute__((vector_size(16)));

__device__ __forceinline__ void async_copy16(const float* g, float* l) {
#if HAVE_ASYNC_LDS
  __builtin_amdgcn_global_load_async_to_lds_b128(
      (__attribute__((address_space(1))) v4i_vs*)(g),
      (__attribute__((address_space(3))) v4i_vs*)(l), 0, 0);
#else
  *(float4*)l = *(const float4*)g;
#endif
}

__device__ __forceinline__ void wait_async_le6() {
#if HAVE_ASYNC_LDS
#if __has_builtin(__builtin_amdgcn_s_wait_asynccnt)
  __builtin_amdgcn_s_wait_asynccnt(6);
#else
  asm volatile("s_wait_asynccnt 0x6" ::: "memory");
#endif
#endif
}

__device__ __forceinline__ void wait_async_le0() {
#if HAVE_ASYNC_LDS
#if __has_builtin(__builtin_amdgcn_s_wait_asynccnt)
  __builtin_amdgcn_s_wait_asynccnt(0);
#else
  asm volatile("s_wait_asynccnt 0x0" ::: "memory");
#endif
#endif
}

// Graph edges (src, dst) from the reference CONN table.
__constant__ int c_src[19] = {15,13,16,14,11,5,6,5,5,6,7,8,1,0,0,1,2,3,4};
__constant__ int c_dst[19] = {13,11,14,12,12,11,12,6,7,8,9,10,2,1,2,3,4,5,6};

// ---------------------------------------------------------------------------
// Kernel 1: fused GCN (layer1 + AHAT agg + relu + layer2(WMMA) + AHAT agg)
// One block per (b, t).  Output feats laid out as (T, B, 2176) for the LSTM.
// ---------------------------------------------------------------------------
__global__ __launch_bounds__(256) void gcn_kernel(
    const float* __restrict__ pose1, const float* __restrict__ pose2,
    const float* __restrict__ W1, const float* __restrict__ b1,
    const float* __restrict__ W2, const float* __restrict__ b2,
    float* __restrict__ xout) {
  __shared__ float sW1[384];
  __shared__ float sB1[128];
  __shared__ float sB2[128];
  __shared__ float sA[289];
  __shared__ float p1s[51];
  __shared__ float p2s[51];
  __shared__ float rbuf[2][32 * 128];  // relu'd layer1, zero-padded to 32 rows
  __shared__ float obuf[2][17 * 128];  // layer1 pre / layer2 out

  const int tid = threadIdx.x;
  const int bt = blockIdx.x;
  const int b = bt >> 8;   // batch   (T = 256)
  const int t = bt & 255;  // time
  const size_t pbase = (size_t)bt * 51;

  for (int i = tid; i < 384; i += 256) sW1[i] = W1[i];
  if (tid < 128) { sB1[tid] = b1[tid]; sB2[tid] = b2[tid]; }
  if (tid < 51) { p1s[tid] = pose1[pbase + tid]; p2s[tid] = pose2[pbase + tid]; }
  if (tid == 0) {  // build normalized adjacency AHAT (17x17)
    float deg[17], dis[17];
    for (int i = 0; i < 17; ++i) deg[i] = 1.0f;  // self loops
    for (int e = 0; e < 19; ++e) deg[c_dst[e]] += 1.0f;
    for (int i = 0; i < 17; ++i) dis[i] = 1.0f / sqrtf(deg[i]);
    for (int i = 0; i < 289; ++i) sA[i] = 0.0f;
    for (int e = 0; e < 19; ++e) {
      int s = c_src[e], d = c_dst[e];
      sA[d * 17 + s] += dis[s] * dis[d];
    }
    for (int i = 0; i < 17; ++i) sA[i * 17 + i] += dis[i] * dis[i];
  }
  __syncthreads();

  // ---- layer 1: g = pose @ W1  (K=3, scalar is optimal) ----
  for (int i = tid; i < 17 * 128; i += 256) {
    int n = i >> 7, h = i & 127;
    obuf[0][i] = p1s[n * 3 + 0] * sW1[h] + p1s[n * 3 + 1] * sW1[128 + h] +
                 p1s[n * 3 + 2] * sW1[256 + h];
    obuf[1][i] = p2s[n * 3 + 0] * sW1[h] + p2s[n * 3 + 1] * sW1[128 + h] +
                 p2s[n * 3 + 2] * sW1[256 + h];
  }
  __syncthreads();

  // ---- AHAT agg (pose1, b==0 only) + bias + relu, pad rows 17..31 with 0 ----
  for (int i = tid; i < 32 * 128; i += 256) {
    int n = i >> 7, h = i & 127;
    float v1 = 0.0f, v2 = 0.0f;
    if (n < 17) {
      float a1;
      if (b == 0) {
        a1 = 0.0f;
        for (int j = 0; j < 17; ++j) a1 += sA[n * 17 + j] * obuf[0][j * 128 + h];
      } else {
        a1 = obuf[0][i];
      }
      v1 = fmaxf(a1 + sB1[h], 0.0f);
      v2 = fmaxf(obuf[1][i] + sB1[h], 0.0f);
    }
    rbuf[0][i] = v1;
    rbuf[1][i] = v2;
  }
  __syncthreads();

  // ---- layer 2: (32x128) @ W2(128x128) via fp32 WMMA; 32 C-tiles / 8 waves ----
  {
    const int wv = tid >> 5, lane = tid & 31;
    const int l16 = lane & 15, koff = (lane >> 4) << 1, hi8 = (lane >> 4) << 3;
#pragma unroll
    for (int tt = 0; tt < 4; ++tt) {
      int tIdx = wv * 4 + tt;
      int pose = tIdx >> 4, mt = (tIdx >> 3) & 1, nt = tIdx & 7;
      const float* rs = rbuf[pose];
      float* os = obuf[pose];
      int m0 = mt * 16, n0 = nt * 16;
      v8f acc = zero8();
      for (int k = 0; k < 128; k += 4) {
        v2f a, bb;
        a.x = rs[(m0 + l16) * 128 + k + koff];
        a.y = rs[(m0 + l16) * 128 + k + koff + 1];
        bb.x = W2[(k + koff) * 128 + n0 + l16];      // B[k][n] = W2[k][n]
        bb.y = W2[(k + koff + 1) * 128 + n0 + l16];
        acc = wmma_f32_16x16x4(a, bb, acc);
      }
#pragma unroll
      for (int q = 0; q < 8; ++q) {
        int row = m0 + q + hi8;
        if (row < 17) os[row * 128 + n0 + l16] = acc[q];
      }
    }
  }
  __syncthreads();

  // ---- second AHAT agg for pose1 (b==0) ----
  for (int i = tid; i < 17 * 128; i += 256) {
    int n = i >> 7, h = i & 127;
    float v;
    if (b == 0) {
      v = 0.0f;
      for (int j = 0; j < 17; ++j) v += sA[n * 17 + j] * obuf[0][j * 128 + h];
    } else {
      v = obuf[0][i];
    }
    rbuf[0][i] = v;
  }
  __syncthreads();

  // feats = 0.5*((h1+b2)+(h2+b2)) stored as x[t][b][n*128+h]
  float* xo = xout + ((size_t)t * 128 + b) * 2176;
  for (int i = tid; i < 2176; i += 256) {
    int h = i & 127;
    xo[i] = 0.5f * (rbuf[0][i] + obuf[1][i]) + sB2[h];
  }
}

// ---------------------------------------------------------------------------
// Kernel 2: input projection GEMM  pre = X(Mx2176) @ W^T(2176x1024) + bih+bhh
// Block tile 128x64, K-tile 32.  Double-buffered LDS fed by CDNA5 async
// global->LDS b128 copies (ASYNCcnt pipelined); K-stride padded to 36 floats
// (144B: 16B-aligned, and 36*l mod 64 is distinct for l=0..15 -> LDS
// bank-conflict-free b64 operand fetches).
// ---------------------------------------------------------------------------
#define KS 36
__global__ __launch_bounds__(256) void proj_kernel(
    const float* __restrict__ X, const float* __restrict__ W,
    const float* __restrict__ bih, const float* __restrict__ bhh,
    float* __restrict__ out) {
  __shared__ float Xs[2][128 * KS];
  __shared__ float Wsh[2][64 * KS];
  const int tid = threadIdx.x;
  const int m0 = blockIdx.x * 128;
  const int n0 = blockIdx.y * 64;
  const int wv = tid >> 5, lane = tid & 31;
  const int l16 = lane & 15, koff = (lane >> 4) << 1, hi8 = (lane >> 4) << 3;

  v8f acc[4];
#pragma unroll
  for (int i = 0; i < 4; ++i) acc[i] = zero8();

  // Issue one K-tile's worth of async copies (6 b128 ops per thread).
  auto stage = [&](int buf, int kk) {
#pragma unroll
    for (int u = 0; u < 4; ++u) {  // X tile: 128x32 = 1024 float4
      int j = tid + 256 * u;
      int r = j >> 3, c4 = j & 7;
      async_copy16(&X[(size_t)(m0 + r) * 2176 + kk + c4 * 4],
                   &Xs[buf][r * KS + c4 * 4]);
    }
#pragma unroll
    for (int u = 0; u < 2; ++u) {  // W tile: 64x32 = 512 float4 (B[k][n]=W[n][k])
      int j = tid + 256 * u;
      int r = j >> 3, c4 = j & 7;
      async_copy16(&W[(size_t)(n0 + r) * 2176 + kk + c4 * 4],
                   &Wsh[buf][r * KS + c4 * 4]);
    }
  };

  stage(0, 0);
  for (int kt = 0; kt < 68; ++kt) {  // 2176 = 68 * 32
    const int cur = kt & 1;
    if (kt + 1 < 68) {
      stage(cur ^ 1, (kt + 1) * 32);  // prefetch next tile
      wait_async_le6();               // only the 6 prefetch ops may remain
    } else {
      wait_async_le0();
    }
    __syncthreads();
    const float* xs = Xs[cur];
    const float* wsh = Wsh[cur];
#pragma unroll
    for (int k = 0; k < 32; k += 4) {
      v2f a;
      a.x = xs[(wv * 16 + l16) * KS + k + koff];
      a.y = xs[(wv * 16 + l16) * KS + k + koff + 1];
#pragma unroll
      for (int nt = 0; nt < 4; ++nt) {
        v2f bb;
        bb.x = wsh[(nt * 16 + l16) * KS + k + koff];
        bb.y = wsh[(nt * 16 + l16) * KS + k + koff + 1];
        acc[nt] = wmma_f32_16x16x4(a, bb, acc[nt]);
      }
    }
    __syncthreads();  // finished reading buf[cur] before it is overwritten
  }
#pragma unroll
  for (int nt = 0; nt < 4; ++nt) {
    int col = n0 + nt * 16 + l16;
    float bias = bih[col] + bhh[col];
#pragma unroll
    for (int q = 0; q < 8; ++q) {
      int row = m0 + wv * 16 + q + hi8;
      out[(size_t)row * 1024 + col] = acc[nt][q] + bias;
    }
  }
}

// ---------------------------------------------------------------------------
// Kernel 3: forward LSTM scan. One workgroup (1024 thr = 32 waves).
// Each wave owns hidden-col tile hc (16 cols) x 4 row tiles; c lives in regs,
// h lives in a global buffer synced with barriers+fences each timestep.
// Gate order processed i, g, f, o so only one accumulator set is live.
// ---------------------------------------------------------------------------
__global__ __launch_bounds__(1024) void lstm_fwd_kernel(
    const float* __restrict__ pre, const float* __restrict__ Whh,
    float* __restrict__ hbuf) {
  const int tid = threadIdx.x;
  const int wv = tid >> 5, lane = tid & 31;
  const int l16 = lane & 15, koff = (lane >> 4) << 1, hi8 = (lane >> 4) << 3;
  const int hc = wv & 15;           // hidden column tile (0..15) -> 16 cols
  const int rbase = (wv >> 4) * 4;  // batch row-tile base (0 or 4)

  for (int i = tid; i < 128 * 256; i += 1024) hbuf[i] = 0.0f;  // h0 = 0
  __threadfence();
  __syncthreads();

  float creg[4][8], si[4][8], cacc[4][8], hreg[4][8];
#pragma unroll
  for (int i = 0; i < 4; ++i)
#pragma unroll
    for (int q = 0; q < 8; ++q) creg[i][q] = 0.0f;  // c0 = 0

#pragma unroll 1
  for (int t = 0; t < 256; ++t) {
#pragma unroll 1
    for (int step = 0; step < 4; ++step) {
      const int gate = (step == 0) ? 0 : (step == 1) ? 2 : (step == 2) ? 1 : 3;
      const int nbase = gate * 256 + hc * 16;
      v8f acc[4];
#pragma unroll
      for (int i = 0; i < 4; ++i) acc[i] = zero8();
#pragma unroll 1
      for (int k = 0; k < 256; k += 4) {
        v2f bb;  // B[k][n] = Whh[n][k]
        bb.x = Whh[(size_t)(nbase + l16) * 256 + k + koff];
        bb.y = Whh[(size_t)(nbase + l16) * 256 + k + koff + 1];
#pragma unroll
        for (int i = 0; i < 4; ++i) {
          v2f a;
          int off = ((rbase + i) * 16 + l16) * 256 + k + koff;
          a.x = hbuf[off];
          a.y = hbuf[off + 1];
          acc[i] = wmma_f32_16x16x4(a, bb, acc[i]);
        }
      }
#pragma unroll
      for (int i = 0; i < 4; ++i) {
#pragma unroll
        for (int q = 0; q < 8; ++q) {
          int row = (rbase + i) * 16 + q + hi8;
          float z = acc[i][q] + pre[((size_t)t * 128 + row) * 1024 + nbase + l16];
          if (step == 0)       si[i][q] = sigmf(z);                         // i
          else if (step == 1)  cacc[i][q] = si[i][q] * tanhf(z);            // g
          else if (step == 2)  creg[i][q] = sigmf(z) * creg[i][q] + cacc[i][q]; // f
          else                 hreg[i][q] = sigmf(z) * tanhf(creg[i][q]);   // o
        }
      }
    }
    __syncthreads();  // all reads of h(t) complete
#pragma unroll
    for (int i = 0; i < 4; ++i)
#pragma unroll
      for (int q = 0; q < 8; ++q) {
        int row = (rbase + i) * 16 + q + hi8;
        hbuf[row * 256 + hc * 16 + l16] = hreg[i][q];
      }
    __threadfence();
    __syncthreads();  // h(t+1) visible
  }
  // hbuf now holds hs_f[-1]
}

// ---------------------------------------------------------------------------
// Kernel 4: backward LSTM collapses to ONE step from zero state (only
// hs_b[-1] is consumed), fused with the relu pooling + concat.
// ---------------------------------------------------------------------------
__global__ __launch_bounds__(256) void tail_kernel(
    const float* __restrict__ hf, const float* __restrict__ pre_b,
    float* __restrict__ out) {
  int idx = blockIdx.x * 256 + threadIdx.x;  // 0 .. 128*256-1
  int b = idx >> 8, j = idx & 255;
  const float* p = pre_b + (size_t)b * 1024;
  float zi = p[j], zg = p[512 + j], zo = p[768 + j];  // z = pre (h0 = 0)
  float c = sigmf(zi) * tanhf(zg);                     // sig(f)*c0 vanishes
  float hb = sigmf(zo) * tanhf(c);
  out[b * 512 + j] = fmaxf(hf[idx], 0.0f);
  out[b * 512 + 256 + j] = fmaxf(hb, 0.0f);
}

// ---------------------------------------------------------------------------
// Kernel 5: classifier  cls = pool(128x512) @ Wc^T(512x625) + bc  via WMMA.
// N padded to 640; B loads select-guarded so EXEC stays all-ones.
// ---------------------------------------------------------------------------
__global__ __launch_bounds__(256) void cls_kernel(
    const float* __restrict__ pool, const float* __restrict__ Wc,
    const float* __restrict__ bc, float* __restrict__ out) {
  const int wv = threadIdx.x >> 5, lane = threadIdx.x & 31;
  const int l16 = lane & 15, koff = (lane >> 4) << 1, hi8 = (lane >> 4) << 3;
  const int n0 = blockIdx.x * 16;
  const int n = n0 + l16;
  const bool valid = (n < 625);
  v8f acc = zero8();
  for (int k = 0; k < 512; k += 4) {
    v2f a, bb;
    a.x = pool[(wv * 16 + l16) * 512 + k + koff];
    a.y = pool[(wv * 16 + l16) * 512 + k + koff + 1];
    bb.x = valid ? Wc[(size_t)n * 512 + k + koff] : 0.0f;
    bb.y = valid ? Wc[(size_t)n * 512 + k + koff + 1] : 0.0f;
    acc = wmma_f32_16x16x4(a, bb, acc);
  }
  if (valid) {
    float bias = bc[n];
#pragma unroll
    for (int q = 0; q < 8; ++q) out[(wv * 16 + q + hi8) * 625 + n] = acc[q] + bias;
  }
}

// ---------------------------------------------------------------------------
extern "C" void kernel_launch(void* const* d_in, const int* in_sizes, int n_in,
                              void* d_out, int out_size, void* d_ws,
                              size_t ws_size, hipStream_t stream) {
  (void)in_sizes; (void)n_in; (void)out_size; (void)ws_size;
  const float* pose1 = (const float*)d_in[0];
  const float* pose2 = (const float*)d_in[1];
  const float* W1 = (const float*)d_in[2];
  const float* b1 = (const float*)d_in[3];
  const float* W2 = (const float*)d_in[4];
  const float* b2 = (const float*)d_in[5];
  const float* Wih_f = (const float*)d_in[6];
  const float* Whh_f = (const float*)d_in[7];
  const float* bih_f = (const float*)d_in[8];
  const float* bhh_f = (const float*)d_in[9];
  const float* Wih_b = (const float*)d_in[10];
  const float* bih_b = (const float*)d_in[12];
  const float* bhh_b = (const float*)d_in[13];
  const float* Wc = (const float*)d_in[14];
  const float* bc = (const float*)d_in[15];
  // d_in[11] (Whh_b) unused: backward scan reduced to one zero-state step.

  float* ws = (float*)d_ws;
  float* feats = ws;                                   // (T,B,2176) = 71,303,168 f
  float* pre_f = feats + (size_t)71303168;             // (T,B,1024) = 33,554,432 f
  float* pre_b = pre_f + (size_t)33554432;             // (B,1024)   =    131,072 f
  float* hbuf = pre_b + (size_t)131072;                // (B,256)    =     32,768 f
  float* outp = (float*)d_out;

  // 1) fused GCN -> feats in (T,B,D) layout
  gcn_kernel<<<dim3(32768), 256, 0, stream>>>(pose1, pose2, W1, b1, W2, b2,
                                              feats);
  // 2) forward input projection (full T) and backward (only t = T-1 needed)
  proj_kernel<<<dim3(256, 16), 256, 0, stream>>>(feats, Wih_f, bih_f, bhh_f,
                                                 pre_f);
  proj_kernel<<<dim3(1, 16), 256, 0, stream>>>(feats + (size_t)255 * 128 * 2176,
                                               Wih_b, bih_b, bhh_b, pre_b);
  // 3) forward LSTM scan (sequential, single workgroup)
  lstm_fwd_kernel<<<dim3(1), 1024, 0, stream>>>(pre_f, Whh_f, hbuf);
  // 4) backward single step + relu pool + concat -> d_out[0:65536]
  tail_kernel<<<dim3(128), 256, 0, stream>>>(hbuf, pre_b, outp);
  // 5) classifier -> d_out[65536:145536]
  cls_kernel<<<dim3(40), 256, 0, stream>>>(outp, Wc, bc, outp + 65536);
}